// GraphSAGE_77214922048048
// MI455X (gfx1250) — compile-verified
//
#include <hip/hip_runtime.h>

// GraphSAGE (2-layer, mean aggregation) for MI455X / gfx1250, wave32.
//
// Pipeline (all on `stream`):
//  1) memsetAsync: zero cnt, agg1, agg2 in workspace
//  2) pack_w_kernel x4: repack weights so WMMA B-fragments are contiguous b64 loads
//  3) deg_kernel:      cnt[dst] += 1 per edge (fp32 atomics, L2-resident)
//  4) scatter_kernel:  agg1[dst][:] += x[src][:]  (float4 gather + 4 atomics per 16B)
//  5) sage_gemm<128,relu>:  h = relu(mean1 @ W1_l + x @ W1_r + b1)   [WMMA f32 16x16x4]
//  6) scatter_kernel:  agg2[dst][:] += h[src][:]
//  7) sage_gemm<64,!relu>:  out = mean2 @ W2_l + h @ W2_r + b2       [WMMA f32 16x16x4]
//  8) log_softmax64:   in-place row-wise log-softmax (wave-per-row, shfl_xor width 32)

typedef __attribute__((ext_vector_type(2))) float v2f;
typedef __attribute__((ext_vector_type(8))) float v8f;

#define F_DIM 128  // feature width of both aggregation stages

__global__ void deg_kernel(const int* __restrict__ dst, float* __restrict__ cnt, int E) {
  int e = blockIdx.x * blockDim.x + threadIdx.x;
  if (e < E) atomicAdd(&cnt[dst[e]], 1.0f);
}

// One thread per (edge, 4-float chunk): E*32 threads total.
__global__ void scatter_kernel(const float* __restrict__ feat,
                               const int* __restrict__ src,
                               const int* __restrict__ dst,
                               float* __restrict__ agg, int E) {
  int tid = blockIdx.x * blockDim.x + threadIdx.x;
  int total = E * 32;
  if (tid >= total) return;
  int e = tid >> 5;
  int c = (tid & 31) << 2;
  int s = src[e];
  int d = dst[e];
  float4 v = *reinterpret_cast<const float4*>(feat + (size_t)s * F_DIM + c);
  float* p = agg + (size_t)d * F_DIM + c;
  atomicAdd(p + 0, v.x);
  atomicAdd(p + 1, v.y);
  atomicAdd(p + 2, v.z);
  atomicAdd(p + 3, v.w);
}

// Repack W[fin][fout] (row-major) -> P[fin/2][fout] of float2 so the WMMA
// B-fragment (K,K+1 at fixed column) is one contiguous 8-byte load.
__global__ void pack_w_kernel(const float* __restrict__ W, float* __restrict__ P,
                              int fin, int fout) {
  int idx = blockIdx.x * blockDim.x + threadIdx.x;
  int total = (fin / 2) * fout;
  if (idx >= total) return;
  int kp = idx / fout;
  int col = idx % fout;
  float2 v = make_float2(W[(size_t)(2 * kp) * fout + col],
                         W[(size_t)(2 * kp + 1) * fout + col]);
  *reinterpret_cast<float2*>(P + 2 * (size_t)idx) = v;
}

// Fused SAGEConv GEMM: out[16 x F_OUT] = (agg/deg) @ Wl + xin @ Wr + bias, optional ReLU.
// One block per 16-row node tile; each wave owns TWO 16-column tiles (32 cols), so each
// A fragment feeds 2 WMMAs. K-loop = 32 steps of V_WMMA_F32_16X16X4_F32.
// Wl/Wr are the PACKED layouts from pack_w_kernel.
template <int F_OUT, bool RELU>
__global__ __launch_bounds__((F_OUT / 32) * 32)
void sage_gemm_kernel(const float* __restrict__ agg, const float* __restrict__ cnt,
                      const float* __restrict__ xin,
                      const float* __restrict__ Wl, const float* __restrict__ Wr,
                      const float* __restrict__ bias,
                      float* __restrict__ out, int n) {
  constexpr int F_IN = 128;
  constexpr int PAD = 4;  // 128+4 stride -> 16 row-reads hit banks 4 apart: conflict-free
  __shared__ float sM[16][F_IN + PAD];  // mean tile
  __shared__ float sX[16][F_IN + PAD];  // self tile
  const int row0 = blockIdx.x * 16;
  const int tid = threadIdx.x;
  constexpr int NTHR = (F_OUT / 32) * 32;

  const int lane = tid & 31;
  const int wave = tid >> 5;
  const int half = lane >> 4;  // 0: K=k0,k0+1 / rows M=0..7 ; 1: K=k0+2,k0+3 / rows M=8..15
  const int lm = lane & 15;
  const int c0 = wave * 32 + lm;        // first column tile
  const int c1 = wave * 32 + 16 + lm;   // second column tile

  // Warm the weight stream for this wave while the LDS staging runs.
  __builtin_prefetch(Wl + 2 * (size_t)(half * F_OUT + c0), 0, 1);
  __builtin_prefetch(Wr + 2 * (size_t)(half * F_OUT + c0), 0, 1);

  // Stage: 16 x 128 floats per matrix, float4-vectorized, mean = agg * 1/max(deg,1).
  for (int i = tid; i < 16 * (F_IN / 4); i += NTHR) {
    int r = i / (F_IN / 4);
    int c4 = (i % (F_IN / 4)) * 4;
    int row = row0 + r;
    int rs = row < n ? row : (n - 1);
    float4 va = *reinterpret_cast<const float4*>(agg + (size_t)rs * F_IN + c4);
    float4 vx = *reinterpret_cast<const float4*>(xin + (size_t)rs * F_IN + c4);
    float inv = 1.0f / fmaxf(cnt[rs], 1.0f);
    sM[r][c4 + 0] = va.x * inv;
    sM[r][c4 + 1] = va.y * inv;
    sM[r][c4 + 2] = va.z * inv;
    sM[r][c4 + 3] = va.w * inv;
    sX[r][c4 + 0] = vx.x;
    sX[r][c4 + 1] = vx.y;
    sX[r][c4 + 2] = vx.z;
    sX[r][c4 + 3] = vx.w;
  }
  __syncthreads();

  v8f acc0 = {};
  v8f acc1 = {};
#pragma unroll 4
  for (int k0 = 0; k0 < F_IN; k0 += 4) {
    const int ka = k0 + 2 * half;   // even
    const int kp = ka >> 1;         // packed-pair row
    // A 16x4 fp32 fragment (ISA layout): lanes 0-15 hold K=k0,k0+1; lanes 16-31 K=k0+2,k0+3
    v2f aM, aX;
    aM.x = sM[lm][ka];
    aM.y = sM[lm][ka + 1];
    aX.x = sX[lm][ka];
    aX.y = sX[lm][ka + 1];
    // B 4x16 fp32 fragments: packed (K,K+1) pair is contiguous -> single b64 load each.
    v2f bL0 = *reinterpret_cast<const v2f*>(Wl + 2 * ((size_t)kp * F_OUT + c0));
    v2f bL1 = *reinterpret_cast<const v2f*>(Wl + 2 * ((size_t)kp * F_OUT + c1));
    v2f bR0 = *reinterpret_cast<const v2f*>(Wr + 2 * ((size_t)kp * F_OUT + c0));
    v2f bR1 = *reinterpret_cast<const v2f*>(Wr + 2 * ((size_t)kp * F_OUT + c1));
    // 8 args: (neg_a, A, neg_b, B, c_mod, C, reuse_a, reuse_b)
    acc0 = __builtin_amdgcn_wmma_f32_16x16x4_f32(false, aM, false, bL0, (short)0, acc0, false, false);
    acc1 = __builtin_amdgcn_wmma_f32_16x16x4_f32(false, aM, false, bL1, (short)0, acc1, false, false);
    acc0 = __builtin_amdgcn_wmma_f32_16x16x4_f32(false, aX, false, bR0, (short)0, acc0, false, false);
    acc1 = __builtin_amdgcn_wmma_f32_16x16x4_f32(false, aX, false, bR1, (short)0, acc1, false, false);
  }

  // C/D layout: lanes 0-15 -> rows 0..7 (VGPR r = row r), lanes 16-31 -> rows 8..15.
  const float bv0 = bias[c0];
  const float bv1 = bias[c1];
  if (row0 + 16 <= n) {
    // Fast path (always taken when N % 16 == 0): branch-free coalesced stores.
#pragma unroll
    for (int r = 0; r < 8; ++r) {
      int row = row0 + half * 8 + r;
      float v0 = acc0[r] + bv0;
      float v1 = acc1[r] + bv1;
      if (RELU) {
        v0 = fmaxf(v0, 0.0f);
        v1 = fmaxf(v1, 0.0f);
      }
      out[(size_t)row * F_OUT + c0] = v0;
      out[(size_t)row * F_OUT + c1] = v1;
    }
  } else {
#pragma unroll
    for (int r = 0; r < 8; ++r) {
      int row = row0 + half * 8 + r;
      if (row < n) {
        float v0 = acc0[r] + bv0;
        float v1 = acc1[r] + bv1;
        if (RELU) {
          v0 = fmaxf(v0, 0.0f);
          v1 = fmaxf(v1, 0.0f);
        }
        out[(size_t)row * F_OUT + c0] = v0;
        out[(size_t)row * F_OUT + c1] = v1;
      }
    }
  }
}

// In-place log-softmax over 64 columns: one 32-lane wave per row, 2 elements/lane.
__global__ void log_softmax64_kernel(float* __restrict__ h, int n) {
  int row = blockIdx.x * (blockDim.x >> 5) + (threadIdx.x >> 5);
  int lane = threadIdx.x & 31;
  if (row >= n) return;
  float* p = h + (size_t)row * 64;
  float a = p[lane];
  float b = p[lane + 32];
  float m = fmaxf(a, b);
#pragma unroll
  for (int o = 16; o > 0; o >>= 1) m = fmaxf(m, __shfl_xor(m, o, 32));
  float s = expf(a - m) + expf(b - m);
#pragma unroll
  for (int o = 16; o > 0; o >>= 1) s += __shfl_xor(s, o, 32);
  float lg = m + logf(s);
  p[lane] = a - lg;
  p[lane + 32] = b - lg;
}

extern "C" void kernel_launch(void* const* d_in, const int* in_sizes, int n_in,
                              void* d_out, int out_size, void* d_ws, size_t ws_size,
                              hipStream_t stream) {
  const float* x = (const float*)d_in[0];    // [N,128]
  const int* ei = (const int*)d_in[1];       // [2,E]
  const float* W1l = (const float*)d_in[2];  // [128,128]
  const float* W1r = (const float*)d_in[3];  // [128,128]
  const float* b1 = (const float*)d_in[4];   // [128]
  const float* W2l = (const float*)d_in[5];  // [128,64]
  const float* W2r = (const float*)d_in[6];  // [128,64]
  const float* b2 = (const float*)d_in[7];   // [64]
  float* out = (float*)d_out;                // [N,64] log-softmax

  const int nnodes = in_sizes[0] / F_DIM;
  const int E = in_sizes[1] / 2;
  const int* src = ei;
  const int* dst = ei + E;

  // Workspace layout: cnt | agg1 | agg2 | h | pW1l | pW1r | pW2l | pW2r
  char* ws = (char*)d_ws;
  const size_t cntBytes = (((size_t)nnodes * sizeof(float)) + 255) & ~(size_t)255;
  const size_t featBytes = (size_t)nnodes * F_DIM * sizeof(float);
  float* cnt = (float*)ws;
  float* agg1 = (float*)(ws + cntBytes);
  float* agg2 = (float*)(ws + cntBytes + featBytes);
  float* h = (float*)(ws + cntBytes + 2 * featBytes);
  float* pW1l = (float*)(ws + cntBytes + 3 * featBytes);
  float* pW1r = pW1l + 128 * 128;
  float* pW2l = pW1r + 128 * 128;
  float* pW2r = pW2l + 128 * 64;

  // Zero cnt + agg1 + agg2 (contiguous) every call; h/out are fully overwritten.
  hipMemsetAsync(ws, 0, cntBytes + 2 * featBytes, stream);

  // Repack weights into WMMA-B-fragment-friendly (K-pair contiguous) layout.
  pack_w_kernel<<<(64 * 128 + 255) / 256, 256, 0, stream>>>(W1l, pW1l, 128, 128);
  pack_w_kernel<<<(64 * 128 + 255) / 256, 256, 0, stream>>>(W1r, pW1r, 128, 128);
  pack_w_kernel<<<(64 * 64 + 255) / 256, 256, 0, stream>>>(W2l, pW2l, 128, 64);
  pack_w_kernel<<<(64 * 64 + 255) / 256, 256, 0, stream>>>(W2r, pW2r, 128, 64);

  deg_kernel<<<(E + 255) / 256, 256, 0, stream>>>(dst, cnt, E);

  const int scatterThreads = E * 32;
  scatter_kernel<<<(scatterThreads + 255) / 256, 256, 0, stream>>>(x, src, dst, agg1, E);

  sage_gemm_kernel<128, true><<<(nnodes + 15) / 16, 128, 0, stream>>>(
      agg1, cnt, x, pW1l, pW1r, b1, h, nnodes);

  scatter_kernel<<<(scatterThreads + 255) / 256, 256, 0, stream>>>(h, src, dst, agg2, E);

  sage_gemm_kernel<64, false><<<(nnodes + 15) / 16, 64, 0, stream>>>(
      agg2, cnt, h, pW2l, pW2r, b2, out, nnodes);

  log_softmax64_kernel<<<(nnodes + 7) / 8, 256, 0, stream>>>(out, nnodes);
}